// GCN_16312285790927
// MI455X (gfx1250) — compile-verified
//
#include <hip/hip_runtime.h>
#include <hip/hip_bf16.h>

typedef float v2f __attribute__((ext_vector_type(2)));
typedef float v8f __attribute__((ext_vector_type(8)));

#define HD 64
#define NG 256

// ---------------------------------------------------------------------------
// Utility: zero a float buffer
// ---------------------------------------------------------------------------
__global__ void zero_f32_kernel(float* __restrict__ p, int n) {
  int i = blockIdx.x * blockDim.x + threadIdx.x;
  if (i < n) p[i] = 0.0f;
}

// ---------------------------------------------------------------------------
// Degree count (deg[i] = #incoming edges), then dis[i] = rsqrt(deg+1) in place
// ---------------------------------------------------------------------------
__global__ void deg_count_kernel(const long long* __restrict__ dst, int E,
                                 float* __restrict__ deg) {
  int i = blockIdx.x * blockDim.x + threadIdx.x;
  if (i < E) atomicAdd(&deg[(int)dst[i]], 1.0f);
}

__global__ void make_dis_kernel(float* __restrict__ deg_dis, int n) {
  int i = blockIdx.x * blockDim.x + threadIdx.x;
  if (i < n) deg_dis[i] = rsqrtf(deg_dis[i] + 1.0f);
}

// ---------------------------------------------------------------------------
// GEMM: h = X[N,64] @ W[64,64] via V_WMMA_F32_16X16X4_F32.
// One wave computes one 16x16 output tile; block = 4 waves = 16 rows x all 64
// cols. Writes hs = dis*h and acc = hs (self-loop pre-seeded for the scatter).
//
// Fragment layouts (CDNA5 ISA 7.12.2, 32-bit data, wave32):
//   A (16x4):  lane L, vgpr v -> M = L%16,          K = 2*(L/16) + v
//   B (4x16):  lane L, vgpr v -> N = L%16,          K = 2*(L/16) + v
//   C (16x16): lane L, vgpr v -> M = v + 8*(L/16),  N = L%16
// ---------------------------------------------------------------------------
__global__ void gcn_gemm_kernel(const float* __restrict__ X,
                                const float* __restrict__ W,
                                const float* __restrict__ dis,
                                float* __restrict__ hs,
                                float* __restrict__ acc, int n_nodes) {
  const int wave = threadIdx.x >> 5;      // 0..3  -> column tile
  const int lane = threadIdx.x & 31;
  const int row0 = blockIdx.x * 16;
  const int col0 = wave * 16;
  const int m  = lane & 15;
  const int hi = lane >> 4;               // 0 or 1

  int row = row0 + m;
  if (row >= n_nodes) row = n_nodes - 1;  // clamp (keeps EXEC uniform)
  const float* __restrict__ xr = X + (size_t)row * HD;

  v8f c = {};
#pragma unroll
  for (int k = 0; k < 16; ++k) {
    const int kk = k * 4 + hi * 2;        // K index of vgpr0 element
    v2f a = *(const v2f*)(xr + kk);       // global_load_b64, 8B aligned
    v2f b;
    b.x = W[(size_t)kk * HD + col0 + m];
    b.y = W[(size_t)(kk + 1) * HD + col0 + m];
    c = __builtin_amdgcn_wmma_f32_16x16x4_f32(
        /*neg_a=*/false, a, /*neg_b=*/false, b,
        /*c_mod=*/(short)0, c, /*reuse_a=*/false, /*reuse_b=*/false);
  }

  // Write back: scale by dis[row] (hs = dis * h); seed acc with hs.
#pragma unroll
  for (int v = 0; v < 8; ++v) {
    const int rr = row0 + v + 8 * hi;
    if (rr < n_nodes) {
      const float val = c[v] * dis[rr];
      const size_t idx = (size_t)rr * HD + col0 + m;
      hs[idx]  = val;
      acc[idx] = val;
    }
  }
}

// ---------------------------------------------------------------------------
// Edge scatter: acc[dst] += hs[src].  One wave32 per edge, float2 per lane.
// ---------------------------------------------------------------------------
__global__ void scatter_kernel(const float* __restrict__ hs,
                               const long long* __restrict__ src,
                               const long long* __restrict__ dst,
                               float* __restrict__ acc, int E) {
  const int t = blockIdx.x * blockDim.x + threadIdx.x;
  const int e = t >> 5;
  const int lane = t & 31;
  if (e >= E) return;

  const int s = (int)src[e];
  const int d = (int)dst[e];

  // prefetch the next wavefront-of-edges' source row (global_prefetch_b8)
  if (lane == 0 && e + 8 < E)
    __builtin_prefetch(hs + (size_t)((int)src[e + 8]) * HD, 0, 0);

  const float2 v = *((const float2*)(hs + (size_t)s * HD) + lane);
  float* ap = acc + (size_t)d * HD + lane * 2;
  atomicAdd(ap,     v.x);
  atomicAdd(ap + 1, v.y);
}

// ---------------------------------------------------------------------------
// Finalize: Y = (relu?)(dis[i]*acc[i,f] + b[f])
// ---------------------------------------------------------------------------
__global__ void finalize_kernel(const float* __restrict__ acc,
                                const float* __restrict__ dis,
                                const float* __restrict__ bias,
                                float* __restrict__ Y, int n, int do_relu) {
  const int t = blockIdx.x * blockDim.x + threadIdx.x;
  if (t >= n * HD) return;
  const int i = t >> 6;
  const int f = t & 63;
  float v = dis[i] * acc[t] + bias[f];
  if (do_relu) v = fmaxf(v, 0.0f);
  Y[t] = v;
}

// ---------------------------------------------------------------------------
// Pool: per-node dot(h3[i], Wl) -> atomic add to sums[batch[i]], count nodes.
// One wave32 per node, float2 per lane, wave reduction via shuffles.
// ---------------------------------------------------------------------------
__global__ void pool_accum_kernel(const float* __restrict__ h,
                                  const long long* __restrict__ batch,
                                  const float* __restrict__ Wl, int n,
                                  float* __restrict__ sums,
                                  float* __restrict__ cnts) {
  const int t = blockIdx.x * blockDim.x + threadIdx.x;
  const int node = t >> 5;
  const int lane = t & 31;
  if (node >= n) return;

  const float2 v = *((const float2*)(h + (size_t)node * HD) + lane);
  float p = v.x * Wl[lane * 2] + v.y * Wl[lane * 2 + 1];
#pragma unroll
  for (int off = 16; off > 0; off >>= 1) p += __shfl_down(p, off, 32);

  if (lane == 0) {
    const int g = (int)batch[node];
    atomicAdd(&sums[g], p);
    atomicAdd(&cnts[g], 1.0f);
  }
}

__global__ void pool_final_kernel(const float* __restrict__ sums,
                                  const float* __restrict__ cnts,
                                  const float* __restrict__ bl,
                                  float* __restrict__ out) {
  const int g = threadIdx.x;
  if (g < NG) out[g] = sums[g] / fmaxf(cnts[g], 1.0f) + bl[0];
}

// ---------------------------------------------------------------------------
// Launch
// ---------------------------------------------------------------------------
extern "C" void kernel_launch(void* const* d_in, const int* in_sizes, int n_in,
                              void* d_out, int out_size, void* d_ws,
                              size_t ws_size, hipStream_t stream) {
  const float*     x     = (const float*)d_in[0];
  const long long* ei    = (const long long*)d_in[1];
  const long long* batch = (const long long*)d_in[2];
  const float* W1 = (const float*)d_in[3];
  const float* b1 = (const float*)d_in[4];
  const float* W2 = (const float*)d_in[5];
  const float* b2 = (const float*)d_in[6];
  const float* W3 = (const float*)d_in[7];
  const float* b3 = (const float*)d_in[8];
  const float* Wl = (const float*)d_in[9];
  const float* bl = (const float*)d_in[10];

  const int N = in_sizes[0] / HD;
  const int E = in_sizes[1] / 2;
  const long long* src = ei;
  const long long* dst = ei + E;

  // workspace layout (floats, 256B-aligned chunks)
  float* dis = (float*)d_ws;
  const size_t nd = ((size_t)N + 63) & ~(size_t)63;
  float* hs   = dis + nd;
  float* acc  = hs  + (size_t)N * HD;
  float* Y    = acc + (size_t)N * HD;
  float* sums = Y   + (size_t)N * HD;
  float* cnts = sums + NG;                 // sums|cnts contiguous (512 floats)

  const dim3 b256(256);

  // degree -> dis (in place)
  zero_f32_kernel<<<(N + 255) / 256, b256, 0, stream>>>(dis, N);
  deg_count_kernel<<<(E + 255) / 256, b256, 0, stream>>>(dst, E, dis);
  make_dis_kernel<<<(N + 255) / 256, b256, 0, stream>>>(dis, N);

  const float* Xin = x;
  const float* Ws[3] = {W1, W2, W3};
  const float* bs[3] = {b1, b2, b3};
  for (int layer = 0; layer < 3; ++layer) {
    gcn_gemm_kernel<<<(N + 15) / 16, dim3(128), 0, stream>>>(
        Xin, Ws[layer], dis, hs, acc, N);
    scatter_kernel<<<(int)(((size_t)E * 32 + 255) / 256), b256, 0, stream>>>(
        hs, src, dst, acc, E);
    finalize_kernel<<<(int)(((size_t)N * HD + 255) / 256), b256, 0, stream>>>(
        acc, dis, bs[layer], Y, N, layer < 2 ? 1 : 0);
    Xin = Y;
  }

  // global mean pool + linear head
  zero_f32_kernel<<<2, b256, 0, stream>>>(sums, 2 * NG);
  pool_accum_kernel<<<(int)(((size_t)N * 32 + 255) / 256), b256, 0, stream>>>(
      Y, batch, Wl, N, sums, cnts);
  pool_final_kernel<<<1, b256, 0, stream>>>(sums, cnts, bl, (float*)d_out);
}